// DFMBPSROIAlign_8400956031314
// MI455X (gfx1250) — compile-verified
//
#include <hip/hip_runtime.h>
#include <hip/hip_bf16.h>
#include <stdint.h>

// ---------------------------------------------------------------------------
// DFMB PSROI-Align for MI455X (gfx1250).
//  - one block = one (ph,pw) bin  x  256 ROIs (8 wave32 waves)
//  - bin's 10 channel planes staged into LDS via CDNA5 per-lane async
//    global->LDS DMA (global_load_async_to_lds_b32, ASYNCcnt), transposed
//    on the fly to pixel-major [1156][12] so each bilinear corner is a
//    contiguous 40B LDS read (ds_load_b128 x2 + ds_load_b64).
//  - gather-throughput bound workload: no dense 16x16xK structure, so the
//    rewarded CDNA5 path is LDS banks + async DMA, not WMMA.
// ---------------------------------------------------------------------------

namespace {
constexpr int kC   = 10;
constexpr int kPH  = 7;
constexpr int kPW  = 7;
constexpr int kH   = 34;
constexpr int kW   = 34;
constexpr int kS   = 4;                 // samples per part
constexpr int kBins = kPH * kPW;        // 49
constexpr int kHW   = kH * kW;          // 1156
constexpr int kPix  = 12;               // padded floats per pixel (48B -> 16B aligned)
constexpr float kInvStride = 1.0f / 8.0f;
}

__global__ __launch_bounds__(256)
void psroi_align_kernel(const float* __restrict__ ft,    // [C*PH*PW, H, W]
                        const float* __restrict__ rois,  // [N, 5]
                        float* __restrict__ out,         // [N, C, 49]
                        int n_rois)
{
    __shared__ float tile[kHW * kPix];                   // 55,488 bytes

    const int tid = threadIdx.x;
    const int bin = blockIdx.y;                          // ph*PW + pw
    const int ph  = bin / kPW;
    const int pw  = bin - ph * kPW;

    // ---- stage + transpose this bin's channel planes into LDS -------------
    // global element (c, bin, p) -> LDS [p][c]; consecutive lanes walk
    // consecutive p within one channel, so the global side stays coalesced
    // while the async unit performs the scatter into the interleaved layout.
    const uint32_t lds_base = (uint32_t)(size_t)&tile[0];   // low 32b = LDS offset
    for (int e = tid; e < kC * kHW; e += 256) {
        const int c = e / kHW;
        const int p = e - c * kHW;
        const float* gptr = ft + ((size_t)(c * kBins + bin) * kHW + p);
        const uint32_t loff = lds_base + (uint32_t)((p * kPix + c) * 4);
        asm volatile("global_load_async_to_lds_b32 %0, %1, off"
                     :: "v"(loff), "v"(gptr)
                     : "memory");
    }
    asm volatile("s_wait_asynccnt 0" ::: "memory");
    __syncthreads();

    const int roi = blockIdx.x * 256 + tid;
    if (roi >= n_rois) return;

    // ---- per-ROI geometry --------------------------------------------------
    const float* r = rois + (size_t)roi * 5;
    const float rsw = r[1] * kInvStride;
    const float rsh = r[2] * kInvStride;
    const float rew = r[3] * kInvStride;
    const float reh = r[4] * kInvStride;

    const float roi_h = fmaxf(reh - rsh, 0.1f);
    const float roi_w = fmaxf(rew - rsw, 0.1f);
    const float bin_h = roi_h * (1.0f / kPH);
    const float bin_w = roi_w * (1.0f / kPW);
    const float sub_h = bin_h * 0.25f;
    const float sub_w = bin_w * 0.25f;
    const float hstart = floorf(rsh + (float)ph * bin_h);
    const float wstart = floorf(rsw + (float)pw * bin_w);

    float acc[kC];
#pragma unroll
    for (int c = 0; c < kC; ++c) acc[c] = 0.0f;
    float count = 0.0f;

#pragma unroll
    for (int ih = 0; ih < kS; ++ih) {
        const float hh  = hstart + ((float)ih + 0.5f) * sub_h;
        const float y1f = floorf(hh);
        const float y2f = ceilf(hh);
        const float dy  = hh - y1f;
        const bool  hok = (hh > -1.0f) && (hh < (float)kH);
        const bool  vy1 = (y1f >= 0.0f) && (y1f < (float)kH);
        const bool  vy2 = (y2f >= 0.0f) && (y2f < (float)kH);
        const int   y1  = (int)fminf(fmaxf(y1f, 0.0f), (float)(kH - 1));
        const int   y2  = (int)fminf(fmaxf(y2f, 0.0f), (float)(kH - 1));

#pragma unroll
        for (int iw = 0; iw < kS; ++iw) {
            const float ww  = wstart + ((float)iw + 0.5f) * sub_w;
            const float x1f = floorf(ww);
            const float x2f = ceilf(ww);
            const float dx  = ww - x1f;
            const bool  wok = (ww > -1.0f) && (ww < (float)kW);
            const float keep = (hok && wok) ? 1.0f : 0.0f;
            count += keep;

            const bool vx1 = (x1f >= 0.0f) && (x1f < (float)kW);
            const bool vx2 = (x2f >= 0.0f) && (x2f < (float)kW);
            const int  x1  = (int)fminf(fmaxf(x1f, 0.0f), (float)(kW - 1));
            const int  x2  = (int)fminf(fmaxf(x2f, 0.0f), (float)(kW - 1));

            const float w11 = keep * ((vy1 && vx1) ? (1.0f - dx) * (1.0f - dy) : 0.0f);
            const float w21 = keep * ((vy2 && vx1) ? (1.0f - dx) * dy          : 0.0f);
            const float w12 = keep * ((vy1 && vx2) ? dx * (1.0f - dy)          : 0.0f);
            const float w22 = keep * ((vy2 && vx2) ? dx * dy                   : 0.0f);

            const float* p11 = &tile[(y1 * kW + x1) * kPix];
            const float* p21 = &tile[(y2 * kW + x1) * kPix];
            const float* p12 = &tile[(y1 * kW + x2) * kPix];
            const float* p22 = &tile[(y2 * kW + x2) * kPix];

#pragma unroll
            for (int c = 0; c < kC; ++c) {
                acc[c] += w11 * p11[c] + w21 * p21[c]
                        + w12 * p12[c] + w22 * p22[c];
            }
        }
    }

    // count==0 implies acc==0, so where(count>0, sum/max(count,1), 0)
    // collapses to a single multiply.
    const float inv = 1.0f / fmaxf(count, 1.0f);
    float* o = out + (size_t)roi * kC * kBins + bin;
#pragma unroll
    for (int c = 0; c < kC; ++c) o[c * kBins] = acc[c] * inv;
}

extern "C" void kernel_launch(void* const* d_in, const int* in_sizes, int n_in,
                              void* d_out, int out_size, void* d_ws, size_t ws_size,
                              hipStream_t stream) {
    const float* ft   = (const float*)d_in[0];   // (1, 490, 34, 34) f32
    const float* rois = (const float*)d_in[1];   // (N, 5) f32
    float* out = (float*)d_out;                  // (N, 10, 49) f32

    const int n = in_sizes[1] / 5;
    dim3 grid((unsigned)((n + 255) / 256), (unsigned)kBins, 1);
    psroi_align_kernel<<<grid, dim3(256, 1, 1), 0, stream>>>(ft, rois, out, n);
}